// FraudGNN_51067161150195
// MI455X (gfx1250) — compile-verified
//
#include <hip/hip_runtime.h>

typedef __attribute__((ext_vector_type(2))) float v2f;
typedef __attribute__((ext_vector_type(8))) float v8f;

#define NNODES 40000
#define NEDGES 640000
#define F 128

// ---------------------------------------------------------------- zero fill
__global__ void zero_f32(float* __restrict__ p, int n) {
  int i = blockIdx.x * blockDim.x + threadIdx.x;
  int stride = gridDim.x * blockDim.x;
  for (; i < n; i += stride) p[i] = 0.0f;
}

// ------------------------------------------------- edge scatter (mean accum)
// one thread per (edge, quad-of-4-features): E * 32 threads.
// float4 gather from feat[src], 4x global_atomic_add_f32 into sum[dst].
__global__ void scatter_accum(const float* __restrict__ feat,
                              const int* __restrict__ ei,
                              float* __restrict__ sum,
                              float* __restrict__ cnt,
                              int do_cnt) {
  int gid = blockIdx.x * blockDim.x + threadIdx.x;
  if (gid >= NEDGES * 32) return;
  int e = gid >> 5;
  int q = gid & 31;
  int src = ei[e];
  int dst = ei[NEDGES + e];
  const float4 v = *(const float4*)(feat + (size_t)src * F + q * 4);
  float* s = sum + (size_t)dst * F + q * 4;
  atomicAdd(s + 0, v.x);
  atomicAdd(s + 1, v.y);
  atomicAdd(s + 2, v.z);
  atomicAdd(s + 3, v.w);
  if (do_cnt && q == 0) atomicAdd(cnt + dst, 1.0f);
}

// --------------------------------------------- fused SAGE layer (WMMA f32)
// out[16 nodes x 128 cols] = relu( mean*Wl + xin*Wr + bias )
// block = 256 threads = 8 waves; wave w owns cols [16w, 16w+16).
// K=128 consumed in steps of 4 via V_WMMA_F32_16X16X4_F32.
__global__ __launch_bounds__(256)
void sage_gemm(const float* __restrict__ sum,
               const float* __restrict__ cnt,
               const float* __restrict__ xin,
               const float* __restrict__ Wl,
               const float* __restrict__ Wr,
               const float* __restrict__ bias,
               float* __restrict__ out,
               int relu) {
  __shared__ float inv[16];
  const int tid  = threadIdx.x;
  const int lane = tid & 31;
  const int wave = tid >> 5;
  const int half = lane >> 4;   // 0: K pair {k,k+1}; 1: K pair {k+2,k+3}
  const int lid  = lane & 15;
  const int base = blockIdx.x * 16;       // node tile start
  const int col  = wave * 16 + lid;       // output feature

  if (tid < 16) {
    float c = cnt[base + tid];
    inv[tid] = 1.0f / fmaxf(c, 1.0f);
  }
  __syncthreads();

  const float* srow = sum + (size_t)(base + lid) * F;
  const float* xrow = xin + (size_t)(base + lid) * F;
  const float scale = inv[lid];

  v8f acc = {};
#pragma unroll 4
  for (int k = 0; k < F; k += 4) {
    const int ka = k + 2 * half;
    // A fragments: row = lid, cols {ka, ka+1}
    v2f a1 = *(const v2f*)(srow + ka);   // neighbor-sum row
    a1 *= scale;                         // -> mean on the fly
    v2f a2 = *(const v2f*)(xrow + ka);   // self features
    // B fragments: rows {ka, ka+1}, col = col  (W is [in, out] row-major)
    v2f b1, b2;
    b1.x = Wl[ka * F + col];
    b1.y = Wl[(ka + 1) * F + col];
    b2.x = Wr[ka * F + col];
    b2.y = Wr[(ka + 1) * F + col];
    acc = __builtin_amdgcn_wmma_f32_16x16x4_f32(false, a1, false, b1,
                                                (short)0, acc, false, false);
    acc = __builtin_amdgcn_wmma_f32_16x16x4_f32(false, a2, false, b2,
                                                (short)0, acc, false, false);
  }

  const float bv = bias[col];
#pragma unroll
  for (int v = 0; v < 8; ++v) {
    // C/D layout: VGPR v holds row (v + 8*half), col = lid within this wave's slice
    int m = base + v + 8 * half;
    float val = acc[v] + bv;
    if (relu) val = fmaxf(val, 0.0f);
    out[(size_t)m * F + col] = val;
  }
}

// ------------------------------------------------------- final FC (128 -> 2)
__global__ void fc_kernel(const float* __restrict__ h,
                          const float* __restrict__ Wfc,
                          const float* __restrict__ bfc,
                          float* __restrict__ out) {
  int node = blockIdx.x * blockDim.x + threadIdx.x;
  if (node >= NNODES) return;
  const float* row = h + (size_t)node * F;
  float a0 = bfc[0], a1 = bfc[1];
#pragma unroll 8
  for (int k = 0; k < F; ++k) {
    float v = row[k];
    a0 += v * Wfc[k * 2 + 0];
    a1 += v * Wfc[k * 2 + 1];
  }
  out[node * 2 + 0] = a0;
  out[node * 2 + 1] = a1;
}

// ---------------------------------------------------------------- launcher
extern "C" void kernel_launch(void* const* d_in, const int* in_sizes, int n_in,
                              void* d_out, int out_size, void* d_ws, size_t ws_size,
                              hipStream_t stream) {
  const float* x   = (const float*)d_in[0];
  const int*   ei  = (const int*)d_in[1];   // [2, E]
  const float* W1l = (const float*)d_in[2];
  const float* b1  = (const float*)d_in[3];
  const float* W1r = (const float*)d_in[4];
  const float* W2l = (const float*)d_in[5];
  const float* b2  = (const float*)d_in[6];
  const float* W2r = (const float*)d_in[7];
  const float* Wfc = (const float*)d_in[8];
  const float* bfc = (const float*)d_in[9];
  float* out = (float*)d_out;

  float* ws     = (float*)d_ws;
  float* sumbuf = ws;                               // N*F
  float* h1     = sumbuf + (size_t)NNODES * F;      // N*F
  float* h2     = h1 + (size_t)NNODES * F;          // N*F
  float* cnt    = h2 + (size_t)NNODES * F;          // N

  const int scatterBlocks = (NEDGES * 32 + 255) / 256;

  // ---- layer 1
  zero_f32<<<2048, 256, 0, stream>>>(sumbuf, NNODES * F);
  zero_f32<<<160, 256, 0, stream>>>(cnt, NNODES);
  scatter_accum<<<scatterBlocks, 256, 0, stream>>>(x, ei, sumbuf, cnt, 1);
  sage_gemm<<<NNODES / 16, 256, 0, stream>>>(sumbuf, cnt, x, W1l, W1r, b1, h1, 1);

  // ---- layer 2 (reuse sum buffer; degree counts unchanged)
  zero_f32<<<2048, 256, 0, stream>>>(sumbuf, NNODES * F);
  scatter_accum<<<scatterBlocks, 256, 0, stream>>>(h1, ei, sumbuf, cnt, 0);
  sage_gemm<<<NNODES / 16, 256, 0, stream>>>(sumbuf, cnt, h1, W2l, W2r, b2, h2, 1);

  // ---- classifier head
  fc_kernel<<<(NNODES + 255) / 256, 256, 0, stream>>>(h2, Wfc, bfc, out);
}